// HomoGNN_75677323755530
// MI455X (gfx1250) — compile-verified
//
#include <hip/hip_runtime.h>

typedef __attribute__((ext_vector_type(2))) float v2f;
typedef __attribute__((ext_vector_type(8))) float v8f;

#define NNODES   50000
#define NEDGES   800000
#define FIN      128
#define HIDF     128
#define OUTF     64
#define NGROUPS  512

// ---------------------------------------------------------------- utilities
__global__ void zero_kernel(float* __restrict__ p, long n) {
  long i = (long)blockIdx.x * blockDim.x + threadIdx.x;
  long stride = (long)gridDim.x * blockDim.x;
  for (; i < n; i += stride) p[i] = 0.0f;
}

__global__ void deg_kernel(const int* __restrict__ col, float* __restrict__ deg, int E) {
  int e = blockIdx.x * blockDim.x + threadIdx.x;
  if (e < E) atomicAdd(&deg[col[e]], 1.0f);
}

__global__ void dis_kernel(const float* __restrict__ deg, float* __restrict__ dis, int n) {
  int i = blockIdx.x * blockDim.x + threadIdx.x;
  if (i < n) {
    float d = deg[i];
    dis[i] = d > 0.0f ? rsqrtf(d) : 0.0f;
  }
}

// ------------------------------------------------ f32 WMMA GEMM: C = A @ B
// A: [M x K] row-major, B: [K x N] row-major, C: [M x N] row-major, N = NT*16.
// Whole B panel is staged in LDS (pair-interleaved, bank-padded) and shared by
// the block's 8 waves; each wave computes a 16-row strip with NT accumulators.
// A is software-pipelined (next k-step's v2f prefetched during the WMMAs).
template<int NT, int K>
__global__ __launch_bounds__(256) void gemm_wmma_f32(const float* __restrict__ A,
                                                     const float* __restrict__ B,
                                                     float* __restrict__ C, int M) {
  constexpr int N = NT * 16;
  constexpr int RS = 2 * N + 32;          // padded dword stride per K-pair row
  __shared__ float Bs[(K / 2) * RS];      // Bs[p*RS + 2n + (k&1)] = B[2p+(k&1)][n]

  const int tid = threadIdx.x;

  // ---- stage B: coalesced b128 global loads, strided LDS scatter (one-time)
  for (int base = tid * 4; base < K * N; base += 256 * 4) {
    const float4 v = *(const float4*)(B + base);
    const int k = base / N;               // N multiple of 4 -> same k for all 4
    const int n = base % N;
    float* dst = &Bs[(k >> 1) * RS + (k & 1)];
    dst[2 * (n + 0)] = v.x;
    dst[2 * (n + 1)] = v.y;
    dst[2 * (n + 2)] = v.z;
    dst[2 * (n + 3)] = v.w;
  }
  __syncthreads();

  const int mt = blockIdx.x * 8 + (tid >> 5);
  if (mt * 16 >= M) return;               // wave-uniform: EXEC all-1s for WMMA
  const int lane = tid & 31;
  const int lm = lane & 15;               // M index (A) / N index (B, C)
  const int kh = lane >> 4;               // lanes 16-31 supply K=2,3 of each step

  const float* __restrict__ arow = A + (size_t)(mt * 16 + lm) * K + 2 * kh;
  const float* bbase = &Bs[kh * RS + 2 * lm];   // pair row p = k0/2 + kh

  v8f acc[NT];
#pragma unroll
  for (int t = 0; t < NT; ++t) acc[t] = (v8f){};

  v2f a = *(const v2f*)(arow);            // K = 2*kh, 2*kh+1
  for (int k0 = 0; k0 < K; k0 += 4) {
    v2f an = a;
    if (k0 + 4 < K) an = *(const v2f*)(arow + k0 + 4);   // prefetch next step
    const float* bk = bbase + (k0 >> 1) * RS;
#pragma unroll
    for (int t = 0; t < NT; ++t) {
      v2f b = *(const v2f*)(bk + 32 * t);                // ds_load_b64
      acc[t] = __builtin_amdgcn_wmma_f32_16x16x4_f32(
          /*neg_a=*/false, a, /*neg_b=*/false, b,
          /*c_mod=*/(short)0, acc[t], /*reuse_a=*/false, /*reuse_b=*/false);
    }
    a = an;
  }

#pragma unroll
  for (int i = 0; i < 8; ++i) {
    const int m = mt * 16 + i + 8 * kh;   // VGPR i: lanes0-15 -> M=i, lanes16-31 -> M=i+8
#pragma unroll
    for (int t = 0; t < NT; ++t) {
      C[(size_t)m * N + t * 16 + lm] = acc[t][i];
    }
  }
}

// --------------------------------------------- edge scatter: dst[col] += norm*src[row]
// One wave per edge; each lane handles VPT consecutive features (vectorized load).
template<int F, int VPT>
__global__ void scatter_edges(const int* __restrict__ row, const int* __restrict__ col,
                              const float* __restrict__ dis, const float* __restrict__ src,
                              float* __restrict__ dst, int E) {
  int gid = blockIdx.x * blockDim.x + threadIdx.x;
  int e = gid >> 5;
  if (e >= E) return;
  int lane = threadIdx.x & 31;
  int r = row[e], c = col[e];
  float nrm = dis[r] * dis[c];
  const float* __restrict__ s = src + (size_t)r * F + lane * VPT;
  float* __restrict__ d = dst + (size_t)c * F + lane * VPT;
  float tmp[VPT];
#pragma unroll
  for (int j = 0; j < VPT; ++j) tmp[j] = s[j];           // merges to b64/b128 load
#pragma unroll
  for (int j = 0; j < VPT; ++j) atomicAdd(&d[j], nrm * tmp[j]);
}

// h = relu(agg + lin + b + c), elementwise over [NNODES, F]; written in place over agg.
__global__ void relu_epilogue_kernel(const float* __restrict__ agg, const float* __restrict__ lin,
                                     const float* __restrict__ b, const float* __restrict__ c,
                                     float* __restrict__ h, long n, int F) {
  long i = (long)blockIdx.x * blockDim.x + threadIdx.x;
  if (i < n) {
    int f = (int)(i & (F - 1));
    float v = agg[i] + lin[i] + b[f] + c[f];
    h[i] = v > 0.0f ? v : 0.0f;
  }
}

// pooled[group[node]] += agg2 + lin2 + b2 + c2
__global__ void pool_kernel(const float* __restrict__ agg, const float* __restrict__ lin,
                            const float* __restrict__ b, const float* __restrict__ c,
                            const int* __restrict__ group, float* __restrict__ pooled, long n) {
  long i = (long)blockIdx.x * blockDim.x + threadIdx.x;
  if (i < n) {
    int node = (int)(i >> 6);             // OUTF == 64
    int f = (int)(i & 63);
    float v = agg[i] + lin[i] + b[f] + c[f];
    atomicAdd(&pooled[(size_t)group[node] * OUTF + f], v);
  }
}

// out[g] = sum_f relu(pooled[g][f]) * Wh[f] + bh
__global__ void head_kernel(const float* __restrict__ pooled, const float* __restrict__ Wh,
                            const float* __restrict__ bh, float* __restrict__ out, int G) {
  int g = blockIdx.x * blockDim.x + threadIdx.x;
  if (g >= G) return;
  const float* p = pooled + (size_t)g * OUTF;
  float s = 0.0f;
#pragma unroll 4
  for (int f = 0; f < OUTF; ++f) {
    float v = p[f];
    v = v > 0.0f ? v : 0.0f;
    s += v * Wh[f];
  }
  out[g] = s + bh[0];
}

// ---------------------------------------------------------------- launcher
extern "C" void kernel_launch(void* const* d_in, const int* in_sizes, int n_in,
                              void* d_out, int out_size, void* d_ws, size_t ws_size,
                              hipStream_t stream) {
  (void)in_sizes; (void)n_in; (void)out_size; (void)ws_size;

  const float* x   = (const float*)d_in[0];
  const int*   ei  = (const int*)d_in[1];     // [2, E] flattened
  const int*   grp = (const int*)d_in[2];
  const float* W1  = (const float*)d_in[3];
  const float* b1  = (const float*)d_in[4];
  const float* L1  = (const float*)d_in[5];
  const float* c1  = (const float*)d_in[6];
  const float* W2  = (const float*)d_in[7];
  const float* b2  = (const float*)d_in[8];
  const float* L2  = (const float*)d_in[9];
  const float* c2  = (const float*)d_in[10];
  const float* Wh  = (const float*)d_in[11];
  const float* bh  = (const float*)d_in[12];
  float* out = (float*)d_out;

  const int* erow = ei;
  const int* ecol = ei + NEDGES;

  // workspace layout (floats)
  float* w = (float*)d_ws;
  float* deg  = w;                                  //   50000
  float* dis  = deg + NNODES;                       //   50000
  float* buf1 = dis + NNODES;                       // 6.40M : xw1; later xw2 (lo half) + agg2 (hi half)
  float* buf2 = buf1 + (size_t)NNODES * HIDF;       // 6.40M : lin1; later lin2
  float* buf3 = buf2 + (size_t)NNODES * HIDF;       // 6.40M : agg1 -> h (in place)
  float* pooled = buf3 + (size_t)NNODES * HIDF;     // 32768
  float* xw1  = buf1;
  float* lin1 = buf2;
  float* agg1 = buf3;
  float* h    = buf3;
  float* xw2  = buf1;                               // 3.20M
  float* agg2 = buf1 + (size_t)NNODES * OUTF;       // 3.20M (upper half of buf1)
  float* lin2 = buf2;

  const int TPB = 256;

  // ---- normalization coefficients
  zero_kernel<<<2048, TPB, 0, stream>>>(deg, (long)NNODES);
  zero_kernel<<<4096, TPB, 0, stream>>>(agg1, (long)NNODES * HIDF);
  zero_kernel<<<64, TPB, 0, stream>>>(pooled, (long)NGROUPS * OUTF);
  deg_kernel<<<(NEDGES + TPB - 1) / TPB, TPB, 0, stream>>>(ecol, deg, NEDGES);
  dis_kernel<<<(NNODES + TPB - 1) / TPB, TPB, 0, stream>>>(deg, dis, NNODES);

  // ---- layer 1 GEMMs: xw1 = x @ W1, lin1 = x @ L1   (50000x128 @ 128x128)
  const int mtiles = NNODES / 16;                   // 3125 exactly
  const int gemm_blocks = (mtiles + 7) / 8;         // 8 waves (m-tiles) per block
  gemm_wmma_f32<HIDF / 16, FIN><<<gemm_blocks, TPB, 0, stream>>>(x, W1, xw1, NNODES);
  gemm_wmma_f32<HIDF / 16, FIN><<<gemm_blocks, TPB, 0, stream>>>(x, L1, lin1, NNODES);

  // ---- layer 1 aggregation + fused epilogue: h = relu(agg1 + b1 + lin1 + c1)
  {
    long thr = (long)NEDGES * 32;
    scatter_edges<HIDF, 4><<<(unsigned)((thr + TPB - 1) / TPB), TPB, 0, stream>>>(
        erow, ecol, dis, xw1, agg1, NEDGES);
    long n = (long)NNODES * HIDF;
    relu_epilogue_kernel<<<(unsigned)((n + TPB - 1) / TPB), TPB, 0, stream>>>(
        agg1, lin1, b1, c1, h, n, HIDF);
  }

  // ---- layer 2 GEMMs: xw2 = h @ W2, lin2 = h @ L2   (50000x128 @ 128x64)
  zero_kernel<<<4096, TPB, 0, stream>>>(agg2, (long)NNODES * OUTF);
  gemm_wmma_f32<OUTF / 16, HIDF><<<gemm_blocks, TPB, 0, stream>>>(h, W2, xw2, NNODES);
  gemm_wmma_f32<OUTF / 16, HIDF><<<gemm_blocks, TPB, 0, stream>>>(h, L2, lin2, NNODES);

  // ---- layer 2 aggregation + pooling
  {
    long thr = (long)NEDGES * 32;
    scatter_edges<OUTF, 2><<<(unsigned)((thr + TPB - 1) / TPB), TPB, 0, stream>>>(
        erow, ecol, dis, xw2, agg2, NEDGES);
    long n = (long)NNODES * OUTF;
    pool_kernel<<<(unsigned)((n + TPB - 1) / TPB), TPB, 0, stream>>>(
        agg2, lin2, b2, c2, grp, pooled, n);
  }

  // ---- head: out = relu(pooled) @ Wh + bh
  head_kernel<<<(NGROUPS + TPB - 1) / TPB, TPB, 0, stream>>>(pooled, Wh, bh, out, NGROUPS);
}